// L_23046794510578
// MI455X (gfx1250) — compile-verified
//
#include <hip/hip_runtime.h>

// Problem constants (from reference)
#define NX0   600
#define NX1   300
#define NVOX  (NX0 * NX1)        // 180000 (NX2 == 1)
#define NPTS  (71 * 32 * 64)     // 145408 frustum points
#define C_CH  128
#define PIXN  2048               // 32*64 pixels per channel slice
#define V4N   (NVOX / 4)         // 45000 float4 groups per channel

typedef __attribute__((ext_vector_type(2))) float v2f;
typedef __attribute__((ext_vector_type(8))) float v8f;

// ---------------------------------------------------------------------------
// Kernel 1: init per-voxel "last point index" buffer to -1 (empty)
// ---------------------------------------------------------------------------
__global__ __launch_bounds__(256) void init_last_kernel(int* __restrict__ last) {
    int tid = blockIdx.x * 256 + threadIdx.x;
    if (tid < NVOX) last[tid] = -1;
}

// ---------------------------------------------------------------------------
// Kernel 2: transform frustum points with V_WMMA_F32_16X16X4_F32 and
// atomicMax the point index into its voxel (== jax segment_max of arange).
//
// One wave handles 16 points per WMMA:
//   A (16x4): rows 0..2 = inv_m rows 0..2, rows 3..15 = 0.
//             ISA layout: lane l holds A[M=l&15][K = (l<16 ? 0,1 : 2,3)].
//   B (4x16): column n = homogeneous point p0+n = [fx*fd*k1, fy*fd*k2, fd, 1].
//             Mirrored layout: lane l holds B[K = (l<16 ? 0,1 : 2,3)][N=l&15].
//   D (16x16): rows 0..2 = geom xyz -> acc[0..2] on lanes 0..15.
// ---------------------------------------------------------------------------
__global__ __launch_bounds__(256) void rank_kernel(const float* __restrict__ frustum,
                                                   const float* __restrict__ invm,
                                                   int* __restrict__ last) {
    const int lane = threadIdx.x & 31;
    const int wave = (blockIdx.x << 3) | (threadIdx.x >> 5); // 8 waves / block
    const int p0   = wave << 4;                              // 16 points / wave
    const int n    = lane & 15;
    const int p    = p0 + n;                                 // this lane's point

    // ---- A operand: transform matrix (rows of inv_m[0:3, :])
    const int klo = (lane < 16) ? 0 : 2;
    v2f a;
    a.x = (n < 3) ? invm[n * 4 + klo]     : 0.0f;
    a.y = (n < 3) ? invm[n * 4 + klo + 1] : 0.0f;

    // ---- B operand: 16 homogeneous frustum points as columns
    const float fx = frustum[3 * p + 0];
    const float fy = frustum[3 * p + 1];
    const float fd = frustum[3 * p + 2];
    const float hx = (fx * fd) * (1700.0f / 512.0f);   // matches (fx*fd)*k order
    const float hy = (fy * fd) * (3517.0f / 1024.0f);
    v2f b;
    b.x = (lane < 16) ? hx : fd;     // K=0 : K=2
    b.y = (lane < 16) ? hy : 1.0f;   // K=1 : K=3

    v8f acc = {};
    // 8 args: (neg_a, A, neg_b, B, c_mod, C, reuse_a, reuse_b)
    acc = __builtin_amdgcn_wmma_f32_16x16x4_f32(
        false, a, false, b, (short)0, acc, false, false);

    // Lanes 0..15 hold D[M=0..7][N=lane] in acc[0..7]; rows 0..2 = geom xyz.
    if (lane < 16) {
        const float gx = acc[0];
        const float gy = acc[1];
        const float gz = acc[2];
        // gidx = trunc((geom - (bx - dx/2)) / dx), matching astype(int32)
        const int ix = (int)((gx + 75.0f) / 0.25f);
        const int iy = (int)((gy +  0.0f) / 0.25f);
        const int iz = (int)((gz + 10.0f) / 20.0f);
        if (ix >= 0 && ix < NX0 && iy >= 0 && iy < NX1 && iz == 0) {
            atomicMax(&last[ix * NX1 + iy], p);
        }
    }
}

// ---------------------------------------------------------------------------
// Kernel 3: produce out[c, vx, vy] = valid ? x[c, pix(last)] : 0
// float4 over voxels: coalesced int4 loads of last[], coalesced float4 stores,
// x gathers stay inside an 8KB per-channel slice (L2-resident, x is 1MB).
// ---------------------------------------------------------------------------
__global__ __launch_bounds__(256) void gather_kernel(const float* __restrict__ x,
                                                     const int*   __restrict__ last,
                                                     float*       __restrict__ out) {
    const int tid = blockIdx.x * 256 + threadIdx.x;   // over C_CH * V4N = 5.76M
    const int c   = tid / V4N;
    const int v4  = tid - c * V4N;

    const int4 lv = ((const int4*)last)[v4];
    const float* __restrict__ xc = x + (c << 11);     // channel slice (2048 floats)

    float4 r;
    r.x = (lv.x >= 0) ? xc[lv.x & (PIXN - 1)] : 0.0f; // pixel = p mod 2048
    r.y = (lv.y >= 0) ? xc[lv.y & (PIXN - 1)] : 0.0f;
    r.z = (lv.z >= 0) ? xc[lv.z & (PIXN - 1)] : 0.0f;
    r.w = (lv.w >= 0) ? xc[lv.w & (PIXN - 1)] : 0.0f;

    ((float4*)out)[tid] = r;
}

// ---------------------------------------------------------------------------
extern "C" void kernel_launch(void* const* d_in, const int* in_sizes, int n_in,
                              void* d_out, int out_size, void* d_ws, size_t ws_size,
                              hipStream_t stream) {
    const float* x       = (const float*)d_in[0];  // (1,1,128,32,64) f32
    const float* frustum = (const float*)d_in[1];  // (71,32,64,3)    f32
    const float* invm    = (const float*)d_in[2];  // (4,4)           f32
    float*       out     = (float*)d_out;          // (1,128,600,300) f32
    int*         last    = (int*)d_ws;             // 180000 ints (720KB scratch)

    (void)in_sizes; (void)n_in; (void)out_size; (void)ws_size;

    // 1) last[v] = -1
    init_last_kernel<<<(NVOX + 255) / 256, 256, 0, stream>>>(last);

    // 2) 145408 points = 9088 wave-groups of 16; 8 waves/block -> 1136 blocks
    rank_kernel<<<(NPTS / 16) / 8, 256, 0, stream>>>(frustum, invm, last);

    // 3) 128 channels * 45000 float4 groups = 5,760,000 threads -> 22500 blocks
    gather_kernel<<<(C_CH * V4N) / 256, 256, 0, stream>>>(x, last, out);
}